// CombineLayer_19078244729115
// MI455X (gfx1250) — compile-verified
//
#include <hip/hip_runtime.h>
#include <hip/hip_bf16.h>

typedef float v2f __attribute__((ext_vector_type(2)));
typedef float v8f __attribute__((ext_vector_type(8)));

#define EMB 128
#define NCHUNK 32  // 128 K-values / 4 per WMMA_F32_16X16X4_F32

// ---------------------------------------------------------------------------
// Kernel 1 (one block, 256 threads):
//   1) v[k] = sum_j Wo[128+j] * Wq[j, k]   (v = Wq^T @ Wo2, 256 floats)
//   2) build per-lane WMMA B fragments for all 32 K-chunks:
//      bfZ[c][lane] = {col0: Wo1 pair, col1: v1 pair, else 0}
//      bfC[c][lane] = {col0: v2 pair, else 0}
//      (col = lane&15 selects the B/D column; lane>>4 selects the K pair)
//   3) init the global min-qid scalar.
// Fragments are identical for every wave of kernel 2 -> branch-free loop.
// ---------------------------------------------------------------------------
__global__ void precompute_v(const float* __restrict__ Wq,
                             const float* __restrict__ Wo,
                             float* __restrict__ v,
                             float* __restrict__ bfZ,   // [NCHUNK*32*2]
                             float* __restrict__ bfC,   // [NCHUNK*32*2]
                             unsigned* __restrict__ minq) {
  const int t = threadIdx.x;  // 0..255
  float s = 0.0f;
#pragma unroll 8
  for (int j = 0; j < EMB; ++j) {
    s = __builtin_fmaf(Wo[EMB + j], Wq[j * (2 * EMB) + t], s);
  }
  v[t] = s;
  if (t == 0) *minq = 0xFFFFFFFFu;
  __syncthreads();

  // 1024 fragment slots (32 chunks x 32 lanes), 4 per thread.
#pragma unroll
  for (int i = 0; i < 4; ++i) {
    const int slot  = t + i * 256;        // [0, 1024)
    const int c     = slot >> 5;          // chunk
    const int lane  = slot & 31;
    const int col   = lane & 15;
    const int k     = 4 * c + ((lane >> 4) << 1);
    float z0v = 0.f, z1v = 0.f, c0v = 0.f, c1v = 0.f;
    if (col == 0) {
      z0v = Wo[k];        z1v = Wo[k + 1];        // column 0 of z-chain: Wo1
      c0v = v[EMB + k];   c1v = v[EMB + k + 1];   // column 0 of z0-chain: v2
    } else if (col == 1) {
      z0v = v[k];         z1v = v[k + 1];         // column 1 of z-chain: v1
    }
    bfZ[slot * 2 + 0] = z0v;  bfZ[slot * 2 + 1] = z1v;
    bfC[slot * 2 + 0] = c0v;  bfC[slot * 2 + 1] = c1v;
  }
}

// ---------------------------------------------------------------------------
// Kernel 2: per-node tables via V_WMMA_F32_16X16X4_F32.
//   aT[n] = dot(z[n],  Wo[0:128])   (D column 0 of z-chain)
//   bT[n] = dot(z[n],  v[0:128])    (D column 1 of z-chain)
//   cT[n] = dot(z0[n], v[128:256])  (D column 0 of z0-chain)
// One wave per 16-node tile, 8 waves per 256-thread block. Inner loop is
// branch-free: 4 coalesced b64 loads + 2 WMMAs per chunk, two interleaved
// accumulator chains to cover the WMMA D->C hazard. Bound by streaming z/z0
// once (~102 MB at 23.3 TB/s).
// ---------------------------------------------------------------------------
__global__ void node_tables(const float* __restrict__ z,
                            const float* __restrict__ z0,
                            const float* __restrict__ bfZ,
                            const float* __restrict__ bfC,
                            float* __restrict__ aT,
                            float* __restrict__ bT,
                            float* __restrict__ cT,
                            int N) {
  const int lane  = threadIdx.x & 31;
  const int wave  = threadIdx.x >> 5;
  const int tile  = blockIdx.x * 8 + wave;
  const int nb    = tile * 16;
  if (nb >= N) return;  // uniform per wave -> EXEC stays all-ones for WMMA

  const int col   = lane & 15;          // A-row (M) for loads; B/D column
  const int khalf = (lane >> 4) << 1;   // which K pair this half-wave owns

  int row = nb + col;
  if (row > N - 1) row = N - 1;         // clamp (tail stores guarded below)
  const float* __restrict__ zr  = z   + (size_t)row * EMB + khalf;
  const float* __restrict__ z0r = z0  + (size_t)row * EMB + khalf;
  const v2f*   __restrict__ bZf = (const v2f*)bfZ + lane;  // stride 32 v2f/chunk
  const v2f*   __restrict__ bCf = (const v2f*)bfC + lane;

  v8f accZ = {0.f, 0.f, 0.f, 0.f, 0.f, 0.f, 0.f, 0.f};
  v8f accC = {0.f, 0.f, 0.f, 0.f, 0.f, 0.f, 0.f, 0.f};

#pragma unroll 4
  for (int c = 0; c < NCHUNK; ++c) {
    v2f aZ = *(const v2f*)(zr  + 4 * c);
    v2f aC = *(const v2f*)(z0r + 4 * c);
    v2f bZ = bZf[c * 32];
    v2f bC = bCf[c * 32];
    accZ = __builtin_amdgcn_wmma_f32_16x16x4_f32(false, aZ, false, bZ,
                                                 (short)0, accZ, false, false);
    accC = __builtin_amdgcn_wmma_f32_16x16x4_f32(false, aC, false, bC,
                                                 (short)0, accC, false, false);
  }

  // D layout: VGPR r = row r (lanes 0-15) / row r+8 (lanes 16-31), lane = col.
  const int mh   = (lane >> 4) * 8;
  const int base = nb + mh;
  if (nb + 16 <= N) {
    // Full tile: wave-uniform fast path, unguarded stores.
    if (col == 0) {
#pragma unroll
      for (int r = 0; r < 8; ++r) { aT[base + r] = accZ[r]; cT[base + r] = accC[r]; }
    } else if (col == 1) {
#pragma unroll
      for (int r = 0; r < 8; ++r) { bT[base + r] = accZ[r]; }
    }
  } else {
    // Tail tile: per-element guards.
    if (col == 0) {
#pragma unroll
      for (int r = 0; r < 8; ++r) {
        if (base + r < N) { aT[base + r] = accZ[r]; cT[base + r] = accC[r]; }
      }
    } else if (col == 1) {
#pragma unroll
      for (int r = 0; r < 8; ++r) {
        if (base + r < N) bT[base + r] = accZ[r];
      }
    }
  }
}

// ---------------------------------------------------------------------------
// Kernel 3: minq = min over edges of max(e0,e1). Deterministic via atomicMin.
// ---------------------------------------------------------------------------
__global__ void edge_min(const long long* __restrict__ e0,
                         const long long* __restrict__ e1,
                         unsigned* __restrict__ minq, int E) {
  __shared__ unsigned smin;
  if (threadIdx.x == 0) smin = 0xFFFFFFFFu;
  __syncthreads();
  unsigned local = 0xFFFFFFFFu;
  for (int i = blockIdx.x * blockDim.x + threadIdx.x; i < E;
       i += gridDim.x * blockDim.x) {
    const long long a0 = e0[i];
    const long long a1 = e1[i];
    const unsigned  u  = (unsigned)(a0 > a1 ? a0 : a1);
    local = (u < local) ? u : local;
  }
  atomicMin(&smin, local);
  __syncthreads();
  if (threadIdx.x == 0) atomicMin(minq, smin);
}

// ---------------------------------------------------------------------------
// Kernel 4: out[e] = aT[e0] + bT[e1] + cT[max(e0,e1)-minq] + bo.
// Streams edge_index (16 MB) + out (4 MB); 1.2 MB tables stay in L2.
// ---------------------------------------------------------------------------
__global__ void edge_out(const long long* __restrict__ e0,
                         const long long* __restrict__ e1,
                         const float* __restrict__ aT,
                         const float* __restrict__ bT,
                         const float* __restrict__ cT,
                         const unsigned* __restrict__ minq,
                         const float* __restrict__ bo,
                         float* __restrict__ out, int E) {
  const int i = blockIdx.x * blockDim.x + threadIdx.x;
  if (i >= E) return;
  const long long a0 = e0[i];
  const long long a1 = e1[i];
  const long long qm = (a0 > a1) ? a0 : a1;
  const unsigned  q  = (unsigned)qm - *minq;
  out[i] = aT[a0] + bT[a1] + cT[q] + bo[0];
}

// ---------------------------------------------------------------------------
extern "C" void kernel_launch(void* const* d_in, const int* in_sizes, int n_in,
                              void* d_out, int out_size, void* d_ws, size_t ws_size,
                              hipStream_t stream) {
  const float*     z  = (const float*)d_in[0];
  const long long* ei = (const long long*)d_in[1];  // int64 per reference
  const float*     z0 = (const float*)d_in[2];
  const float*     Wq = (const float*)d_in[3];
  const float*     Wo = (const float*)d_in[4];
  const float*     bo = (const float*)d_in[5];

  const int N = in_sizes[0] / EMB;   // z: [N, 128]
  const int E = in_sizes[1] / 2;     // edge_index: [2, E]
  const long long* e0 = ei;
  const long long* e1 = ei + E;

  // Workspace: aT[N] | bT[N] | cT[N] | v[256] | bfZ[2048] | bfC[2048] | minq
  float*    aT   = (float*)d_ws;
  float*    bT   = aT + N;
  float*    cT   = bT + N;
  float*    v    = cT + N;
  float*    bfZ  = v + 2 * EMB;
  float*    bfC  = bfZ + NCHUNK * 32 * 2;
  unsigned* minq = (unsigned*)(bfC + NCHUNK * 32 * 2);

  hipLaunchKernelGGL(precompute_v, dim3(1), dim3(256), 0, stream,
                     Wq, Wo, v, bfZ, bfC, minq);

  const int nTiles  = (N + 15) / 16;
  const int nBlocks = (nTiles + 7) / 8;  // 8 waves (tiles) per 256-thread block
  hipLaunchKernelGGL(node_tables, dim3(nBlocks), dim3(256), 0, stream,
                     z, z0, bfZ, bfC, aT, bT, cT, N);

  hipLaunchKernelGGL(edge_min, dim3(1024), dim3(256), 0, stream, e0, e1, minq, E);

  hipLaunchKernelGGL(edge_out, dim3((E + 255) / 256), dim3(256), 0, stream,
                     e0, e1, aT, bT, cT, minq, bo, (float*)d_out, E);
}